// GCNEncoder_65807488910096
// MI455X (gfx1250) — compile-verified
//
#include <hip/hip_runtime.h>

typedef __attribute__((ext_vector_type(2))) float v2f;
typedef __attribute__((ext_vector_type(8))) float v8f;

#define N_NODES 100000
#define N_EDGES 1600000
#define IN_CH   128
#define FEAT    64   // HID == OUT == 64

// ---------------------------------------------------------------------------
// Utility: zero-fill a float buffer
// ---------------------------------------------------------------------------
__global__ void fill_zero_kernel(float* __restrict__ p, long long n) {
    long long i = (long long)blockIdx.x * blockDim.x + threadIdx.x;
    if (i < n) p[i] = 0.0f;
}

// ---------------------------------------------------------------------------
// Degree: deg[d] += 1 per edge (self-loop +1 folded into dinv kernel)
// ---------------------------------------------------------------------------
__global__ void degree_kernel(const long long* __restrict__ dst,
                              float* __restrict__ deg) {
    int e = blockIdx.x * blockDim.x + threadIdx.x;
    if (e < N_EDGES) {
        atomicAdd(&deg[(int)dst[e]], 1.0f);
    }
}

__global__ void dinv_kernel(float* __restrict__ deg) {
    int n = blockIdx.x * blockDim.x + threadIdx.x;
    if (n < N_NODES) {
        deg[n] = rsqrtf(deg[n] + 1.0f);  // D^{-1/2} with self-loop
    }
}

// ---------------------------------------------------------------------------
// H[N,64] = A[N,K] @ W[K,64], f32, row-major, via V_WMMA_F32_16X16X4_F32.
// One wave computes a 16-row stripe (4 column tiles of 16), K-loop step 4.
//
// A-frag (16x4 f32, ISA 7.12.2): lanes 0-15 hold K=0,1 ; lanes 16-31 K=2,3,
// row M = lane%16 per VGPR pair.
// B-frag (4x16): VGPR0 lanes0-15 = row K=0, lanes16-31 = row K=2;
//                VGPR1 lanes0-15 = row K=1, lanes16-31 = row K=3; col = lane%16.
// C/D (16x16 f32): VGPR r, lane l -> row = r + 8*(l/16), col = l%16.
// ---------------------------------------------------------------------------
__global__ __launch_bounds__(256) void gemm_wmma_kernel(
    const float* __restrict__ A, const float* __restrict__ W,
    float* __restrict__ H, int K)
{
    const int wave = blockIdx.x * (blockDim.x >> 5) + (threadIdx.x >> 5);
    const int lane = threadIdx.x & 31;
    const int row0 = wave * 16;
    if (row0 >= N_NODES) return;         // uniform per wave: EXEC stays all-1s

    const int half = lane >> 4;          // 0: K=0,1  1: K=2,3
    const int lm   = lane & 15;          // row (A) / col (B,C)

    v8f acc0 = (v8f)0.0f, acc1 = (v8f)0.0f, acc2 = (v8f)0.0f, acc3 = (v8f)0.0f;

    const float* arow = A + (long long)(row0 + lm) * K;

    for (int k0 = 0; k0 < K; k0 += 4) {
        const int ka = k0 + 2 * half;
        v2f a;
        a.x = arow[ka];
        a.y = arow[ka + 1];

        const float* w0 = W + (long long)ka * FEAT + lm;        // row ka
        const float* w1 = W + (long long)(ka + 1) * FEAT + lm;  // row ka+1

        v2f b0; b0.x = w0[0];  b0.y = w1[0];
        v2f b1; b1.x = w0[16]; b1.y = w1[16];
        v2f b2; b2.x = w0[32]; b2.y = w1[32];
        v2f b3; b3.x = w0[48]; b3.y = w1[48];

        acc0 = __builtin_amdgcn_wmma_f32_16x16x4_f32(false, a, false, b0, (short)0, acc0, false, false);
        acc1 = __builtin_amdgcn_wmma_f32_16x16x4_f32(false, a, false, b1, (short)0, acc1, false, false);
        acc2 = __builtin_amdgcn_wmma_f32_16x16x4_f32(false, a, false, b2, (short)0, acc2, false, false);
        acc3 = __builtin_amdgcn_wmma_f32_16x16x4_f32(false, a, false, b3, (short)0, acc3, false, false);
    }

    float* hrow = H + (long long)row0 * FEAT + lm;
#pragma unroll
    for (int r = 0; r < 8; ++r) {
        const long long off = (long long)(r + 8 * half) * FEAT;
        hrow[off +  0] = acc0[r];
        hrow[off + 16] = acc1[r];
        hrow[off + 32] = acc2[r];
        hrow[off + 48] = acc3[r];
    }
}

// ---------------------------------------------------------------------------
// Edge aggregation: agg[dst] += h[src] * dinv[src]*dinv[dst]
// 64 threads per edge (one channel each): coalesced 256B gather + 256B atomics.
// ---------------------------------------------------------------------------
__global__ void aggregate_kernel(const float* __restrict__ h,
                                 const long long* __restrict__ src,
                                 const long long* __restrict__ dst,
                                 const float* __restrict__ dinv,
                                 float* __restrict__ agg)
{
    long long t = (long long)blockIdx.x * blockDim.x + threadIdx.x;
    long long e = t >> 6;
    int c = (int)(t & 63);
    if (e >= N_EDGES) return;
    int s = (int)src[e];
    int d = (int)dst[e];
    float norm = dinv[s] * dinv[d];
    atomicAdd(&agg[d * FEAT + c], h[s * FEAT + c] * norm);
}

// ---------------------------------------------------------------------------
// Layer-1 epilogue (in place on agg): self-loop + bias + ReLU + dropout
// ---------------------------------------------------------------------------
__global__ void finalize1_kernel(float* __restrict__ agg,
                                 const float* __restrict__ h,
                                 const float* __restrict__ dinv,
                                 const float* __restrict__ b,
                                 const float* __restrict__ mask)
{
    long long t = (long long)blockIdx.x * blockDim.x + threadIdx.x;
    if (t >= (long long)N_NODES * FEAT) return;
    int n = (int)(t >> 6);
    int c = (int)(t & 63);
    float di = dinv[n];
    float v = agg[t] + h[t] * di * di + b[c];
    v = fmaxf(v, 0.0f) * mask[t];
    agg[t] = v;
}

// ---------------------------------------------------------------------------
// Layer-2 epilogue (in place on out): self-loop + bias
// ---------------------------------------------------------------------------
__global__ void finalize2_kernel(float* __restrict__ out,
                                 const float* __restrict__ h,
                                 const float* __restrict__ dinv,
                                 const float* __restrict__ b)
{
    long long t = (long long)blockIdx.x * blockDim.x + threadIdx.x;
    if (t >= (long long)N_NODES * FEAT) return;
    int n = (int)(t >> 6);
    int c = (int)(t & 63);
    float di = dinv[n];
    out[t] = out[t] + h[t] * di * di + b[c];
}

// ---------------------------------------------------------------------------
extern "C" void kernel_launch(void* const* d_in, const int* in_sizes, int n_in,
                              void* d_out, int out_size, void* d_ws, size_t ws_size,
                              hipStream_t stream) {
    const float*      x     = (const float*)d_in[0];                 // [N,128]
    const long long*  eidx  = (const long long*)d_in[1];             // [2,E] int64
    const float*      W1    = (const float*)d_in[2];                 // [128,64]
    const float*      b1    = (const float*)d_in[3];                 // [64]
    const float*      W2    = (const float*)d_in[4];                 // [64,64]
    const float*      b2    = (const float*)d_in[5];                 // [64]
    const float*      mask  = (const float*)d_in[6];                 // [N,64]
    float*            out   = (float*)d_out;                         // [N,64]

    const long long* src = eidx;
    const long long* dst = eidx + N_EDGES;

    // Workspace layout (floats)
    float* ws    = (float*)d_ws;
    float* dinv  = ws;                                   // N
    float* h1    = dinv + N_NODES;                       // N*64
    float* agg1  = h1   + (long long)N_NODES * FEAT;     // N*64 (becomes h_relu)
    float* h2    = agg1 + (long long)N_NODES * FEAT;     // N*64

    const int BT = 256;
    const long long NF = (long long)N_NODES * FEAT;
    const int g_nodes  = (N_NODES + BT - 1) / BT;
    const int g_edges  = (N_EDGES + BT - 1) / BT;
    const int g_nf     = (int)((NF + BT - 1) / BT);
    const int g_ef     = (int)(((long long)N_EDGES * FEAT + BT - 1) / BT);
    const int g_gemm   = (N_NODES / 16 + (BT / 32) - 1) / (BT / 32);

    // Degrees -> dinv
    fill_zero_kernel<<<g_nodes, BT, 0, stream>>>(dinv, N_NODES);
    degree_kernel<<<g_edges, BT, 0, stream>>>(dst, dinv);
    dinv_kernel<<<g_nodes, BT, 0, stream>>>(dinv);

    // Layer 1
    gemm_wmma_kernel<<<g_gemm, BT, 0, stream>>>(x, W1, h1, IN_CH);
    fill_zero_kernel<<<g_nf, BT, 0, stream>>>(agg1, NF);
    aggregate_kernel<<<g_ef, BT, 0, stream>>>(h1, src, dst, dinv, agg1);
    finalize1_kernel<<<g_nf, BT, 0, stream>>>(agg1, h1, dinv, b1, mask);

    // Layer 2
    gemm_wmma_kernel<<<g_gemm, BT, 0, stream>>>(agg1, W2, h2, FEAT);
    fill_zero_kernel<<<g_nf, BT, 0, stream>>>(out, NF);
    aggregate_kernel<<<g_ef, BT, 0, stream>>>(h2, src, dst, dinv, out);
    finalize2_kernel<<<g_nf, BT, 0, stream>>>(out, h2, dinv, b2);
}